// RGAPConv_17995912970446
// MI455X (gfx1250) — compile-verified
//
#include <hip/hip_runtime.h>
#include <hip/hip_bf16.h>

#define D 128
#define NTYPES 16

// fixed small-region float offsets inside d_ws
#define WS_WVM    0          // 128*128 (pair-interleaved: [k/2][n][2])
#define WS_BVM    16384      // 128
#define WS_RELM   16512      // 16*128
#define WS_RELA3  18560      // 16
#define WS_WKA1   18576      // 128
#define WS_WQA2   18704      // 128
#define WS_WKG    18832      // 128
#define WS_WQG    18960      // 128
#define WS_CONST  19088      // 4
#define WS_BIG    19200      // node-sized arrays start here

typedef float v2f __attribute__((ext_vector_type(2)));
typedef float v8f __attribute__((ext_vector_type(8)));

__device__ __forceinline__ float lrelu02(float v) { return v > 0.f ? v : 0.2f * v; }
__device__ __forceinline__ float sigmoidf(float v) { return 1.f / (1.f + __expf(-v)); }

__device__ __forceinline__ void atomicMaxFloatBits(int* addr, float val) {
    if (val >= 0.f) atomicMax(addr, __float_as_int(val));
    else            atomicMin((unsigned int*)addr, __float_as_uint(val));
}

// ---------------- init: zero out-accumulator, init segment max/sum -------------
__global__ void init_kernel(float* __restrict__ out, int* __restrict__ segmax,
                            float* __restrict__ segsum, int nTot, int nNodes) {
    int i = blockIdx.x * blockDim.x + threadIdx.x;
    if (i < nTot) out[i] = 0.f;
    if (i < nNodes) { segmax[i] = 0xFF800000; segsum[i] = 0.f; }  // -inf bits, 0
}

// ---------------- precompute folded weights (1 block, 128 threads) -------------
__global__ void precompute_kernel(const float* __restrict__ Wq, const float* __restrict__ bq,
                                  const float* __restrict__ Wk, const float* __restrict__ bk,
                                  const float* __restrict__ Wv, const float* __restrict__ bv,
                                  const float* __restrict__ rel_emb, const float* __restrict__ attn_vec,
                                  const float* __restrict__ gnw, const float* __restrict__ msg_w,
                                  const float* __restrict__ msg_b, float* __restrict__ ws) {
    int j = threadIdx.x;  // 0..127 == output column n
    // Wvm[k, j] = (Wv @ msg_w)[k, j], stored pair-interleaved for b64 WMMA feeds:
    //   ws[WS_WVM + (k/2)*256 + j*2 + (k&1)]
    for (int k = 0; k < D; ++k) {
        float s = 0.f;
        for (int i = 0; i < D; ++i) s += Wv[k * D + i] * msg_w[i * D + j];
        ws[WS_WVM + (k >> 1) * (2 * D) + j * 2 + (k & 1)] = s;
    }
    // bvm[j] = bv @ msg_w[:, j] + msg_b[j]
    {
        float s = msg_b[j];
        for (int k = 0; k < D; ++k) s += bv[k] * msg_w[k * D + j];
        ws[WS_BVM + j] = s;
    }
    // relM[t, j] = rel_emb[t] @ msg_w[:, j]
    for (int t = 0; t < NTYPES; ++t) {
        float s = 0.f;
        for (int k = 0; k < D; ++k) s += rel_emb[t * D + k] * msg_w[k * D + j];
        ws[WS_RELM + t * D + j] = s;
    }
    // folded attention / gate vectors (row dot of weight matrices)
    {
        float s1 = 0.f, s2 = 0.f, s3 = 0.f, s4 = 0.f;
        for (int k = 0; k < D; ++k) {
            float wkj = Wk[j * D + k], wqj = Wq[j * D + k];
            s1 += wkj * attn_vec[k];          // Wk @ a1
            s2 += wqj * attn_vec[D + k];      // Wq @ a2
            s3 += wkj * gnw[k];               // Wk @ gnw[:d]
            s4 += wqj * gnw[D + k];           // Wq @ gnw[d:]
        }
        ws[WS_WKA1 + j] = s1; ws[WS_WQA2 + j] = s2;
        ws[WS_WKG + j] = s3;  ws[WS_WQG + j] = s4;
    }
    if (j < NTYPES) {  // rel_a3[t] = rel_emb[t] @ a3
        float s = 0.f;
        for (int k = 0; k < D; ++k) s += rel_emb[j * D + k] * attn_vec[2 * D + k];
        ws[WS_RELA3 + j] = s;
    }
    if (j == 0) {  // bias constants
        float c0 = 0.f, c1 = 0.f, c2 = 0.f, c3 = 0.f;
        for (int k = 0; k < D; ++k) {
            c0 += bk[k] * attn_vec[k];
            c1 += bq[k] * attn_vec[D + k];
            c2 += bk[k] * gnw[k];
            c3 += bq[k] * gnw[D + k];
        }
        ws[WS_CONST + 0] = c0; ws[WS_CONST + 1] = c1;
        ws[WS_CONST + 2] = c2; ws[WS_CONST + 3] = c3;
    }
}

// ------- VM = x @ Wvm + bvm via V_WMMA_F32_16X16X4_F32, fused node scalars -----
// block = 256 threads = 8 waves; each block: one 16-row tile; wave w -> 16 cols.
// After the GEMM, wave w computes the 4 folded per-node scalars for tile rows
// 2w, 2w+1 straight from the LDS x-tile (saves a full second pass over x).
__global__ __launch_bounds__(256)
void vm_gemm_kernel(const float* __restrict__ x, const float* __restrict__ ws,
                    float* __restrict__ VM,
                    float* __restrict__ s1, float* __restrict__ s2,
                    float* __restrict__ g1, float* __restrict__ g2, int nNodes) {
    __shared__ float xs[16 * 132];  // +4 pad: conflict-free column reads
    const float* Wp  = ws + WS_WVM;  // pair-interleaved [k/2][n][2]
    const float* bvm = ws + WS_BVM;
    int m0 = blockIdx.x * 16;
    int t = threadIdx.x;
    {   // stage 16x128 x-tile, coalesced float4 pairs (rows clamped for safety)
        int row = t >> 4;
        int col = (t & 15) * 8;
        int grow = m0 + row; if (grow >= nNodes) grow = nNodes - 1;
        const float4* s4 = reinterpret_cast<const float4*>(x + (size_t)grow * D + col);
        float4* d4 = reinterpret_cast<float4*>(xs + row * 132 + col);
        d4[0] = s4[0]; d4[1] = s4[1];
    }
    __syncthreads();
    int lane = t & 31;
    int wv = t >> 5;
    int n0 = wv * 16;
    int lm = lane & 15;
    int kk = (lane >> 4) << 1;   // 0 for lanes 0-15, 2 for lanes 16-31
    v8f acc;
    float bias = bvm[n0 + lm];
    #pragma unroll
    for (int i = 0; i < 8; ++i) acc[i] = bias;
    const float* xrow  = xs + lm * 132 + kk;                       // A row feed
    const float* bbase = Wp + (kk >> 1) * (2 * D) + (n0 + lm) * 2; // B pair feed
    #pragma unroll
    for (int k0 = 0; k0 < D; k0 += 4) {
        v2f a = *reinterpret_cast<const v2f*>(xrow + k0);          // ds b64
        v2f b = *reinterpret_cast<const v2f*>(bbase + (k0 >> 1) * (2 * D)); // global b64
        acc = __builtin_amdgcn_wmma_f32_16x16x4_f32(false, a, false, b,
                                                    (short)0, acc, false, false);
    }
    int col = n0 + lm;
    int rbase = m0 + ((lane >> 4) << 3);           // lanes>=16 hold M=i+8
    #pragma unroll
    for (int i = 0; i < 8; ++i) {
        int r = rbase + i;
        if (r < nNodes) VM[(size_t)r * D + col] = acc[i];
    }

    // ---- fused per-node folded scalars from the staged x-tile ----
    const float* wka1 = ws + WS_WKA1;
    const float* wqa2 = ws + WS_WQA2;
    const float* wkg  = ws + WS_WKG;
    const float* wqg  = ws + WS_WQG;
    #pragma unroll
    for (int rr = 0; rr < 2; ++rr) {
        int r = wv * 2 + rr;
        const float* xr = xs + r * 132;
        float a = 0.f, b = 0.f, c = 0.f, d = 0.f;
        #pragma unroll
        for (int j = 0; j < D; j += 32) {
            float xv = xr[j + lane];
            a += xv * wka1[j + lane];
            b += xv * wqa2[j + lane];
            c += xv * wkg[j + lane];
            d += xv * wqg[j + lane];
        }
        #pragma unroll
        for (int off = 16; off > 0; off >>= 1) {
            a += __shfl_xor(a, off); b += __shfl_xor(b, off);
            c += __shfl_xor(c, off); d += __shfl_xor(d, off);
        }
        int node = m0 + r;
        if (lane == 0 && node < nNodes) {
            s1[node] = a + ws[WS_CONST + 0];
            s2[node] = b + ws[WS_CONST + 1];
            g1[node] = c + ws[WS_CONST + 2];
            g2[node] = d + ws[WS_CONST + 3];
        }
    }
}

// ---------------- per-edge logits, gate, rule MLP, segment max -----------------
__global__ __launch_bounds__(256)
void edge_kernel(const int* __restrict__ ei, const int* __restrict__ et,
                 const float* __restrict__ erf,
                 const float* __restrict__ rule_w1, const float* __restrict__ rule_b1,
                 const float* __restrict__ rule_w2, const float* __restrict__ rule_b2,
                 const float* __restrict__ gate_rule_w, const float* __restrict__ gate_rule_b,
                 const float* __restrict__ gnb,
                 const float* __restrict__ ws,
                 const float* __restrict__ s1, const float* __restrict__ s2,
                 const float* __restrict__ g1, const float* __restrict__ g2,
                 float* __restrict__ e_buf, float* __restrict__ out_gamma,
                 float* __restrict__ out_bij, int* __restrict__ segmax, int nE) {
    int e = blockIdx.x * blockDim.x + threadIdx.x;
    if (e >= nE) return;
    int src = ei[e], dst = ei[nE + e], ty = et[e];
    float f0 = erf[e * 4 + 0], f1 = erf[e * 4 + 1], f2 = erf[e * 4 + 2], f3 = erf[e * 4 + 3];
    // e_base = leaky_relu(K_src.a1 + Q_dst.a2 + rel.a3)
    float e_base = lrelu02(s1[src] + s2[dst] + ws[WS_RELA3 + ty]);
    // b_ij = relu(f @ rule_w1 + rule_b1) @ rule_w2 + rule_b2
    float bij = 0.f;
    #pragma unroll 4
    for (int j = 0; j < D; ++j) {
        float h = f0 * rule_w1[j] + f1 * rule_w1[D + j] + f2 * rule_w1[2 * D + j]
                + f3 * rule_w1[3 * D + j] + rule_b1[j];
        bij += fmaxf(h, 0.f) * rule_w2[j];
    }
    bij += rule_b2[0];
    // gate
    float gl = f0 * gate_rule_w[0] + f1 * gate_rule_w[1] + f2 * gate_rule_w[2]
             + f3 * gate_rule_w[3] + gate_rule_b[0] + g1[src] + g2[dst] + gnb[0];
    float gamma = sigmoidf(gl);
    float ev = e_base + gamma * bij;
    out_gamma[e] = gamma;
    out_bij[e] = bij;
    e_buf[e] = ev;
    atomicMaxFloatBits(&segmax[dst], ev);
}

// ---------------- e_exp + segment sum ------------------------------------------
__global__ __launch_bounds__(256)
void exp_sum_kernel(const int* __restrict__ ei, const int* __restrict__ segmax,
                    float* __restrict__ e_buf, float* __restrict__ segsum, int nE) {
    int e = blockIdx.x * blockDim.x + threadIdx.x;
    if (e >= nE) return;
    int dst = ei[nE + e];
    float mx = __int_as_float(segmax[dst]);
    float ee = __expf(e_buf[e] - mx);
    e_buf[e] = ee;           // overwrite e with e_exp
    atomicAdd(&segsum[dst], ee);
}

// ---------------- alpha + scattered message aggregation (1 wave / edge) --------
__global__ __launch_bounds__(256)
void alpha_msg_kernel(const int* __restrict__ ei, const int* __restrict__ et,
                      const float* __restrict__ VM, const float* __restrict__ ws,
                      const float* __restrict__ segsum,
                      float* __restrict__ e_alpha, float* __restrict__ out, int nE) {
    int e = blockIdx.x * 8 + (threadIdx.x >> 5);
    int lane = threadIdx.x & 31;
    if (e >= nE) return;
    int src = ei[e], dst = ei[nE + e], ty = et[e];
    float alpha = e_alpha[e] / (segsum[dst] + 1e-16f);   // e_alpha currently = e_exp
    const float* vm = VM + (size_t)src * D;
    const float* rm = ws + WS_RELM + ty * D;
    float* od = out + (size_t)dst * D;
    #pragma unroll
    for (int j = 0; j < D; j += 32) {
        int c = j + lane;
        atomicAdd(&od[c], (vm[c] + rm[c]) * alpha);
    }
    if (lane == 0) e_alpha[e] = alpha;                   // finalize output slot
}

extern "C" void kernel_launch(void* const* d_in, const int* in_sizes, int n_in,
                              void* d_out, int out_size, void* d_ws, size_t ws_size,
                              hipStream_t stream) {
    const float* x          = (const float*)d_in[0];
    const int*   edge_index = (const int*)d_in[1];
    const int*   edge_type  = (const int*)d_in[2];
    const float* erf        = (const float*)d_in[3];
    const float* Wq  = (const float*)d_in[5];  const float* bq  = (const float*)d_in[6];
    const float* Wk  = (const float*)d_in[7];  const float* bk  = (const float*)d_in[8];
    const float* Wv  = (const float*)d_in[9];  const float* bv  = (const float*)d_in[10];
    const float* rel_emb  = (const float*)d_in[11];
    const float* attn_vec = (const float*)d_in[12];
    const float* rule_w1 = (const float*)d_in[13]; const float* rule_b1 = (const float*)d_in[14];
    const float* rule_w2 = (const float*)d_in[15]; const float* rule_b2 = (const float*)d_in[16];
    const float* gate_rule_w = (const float*)d_in[17]; const float* gate_rule_b = (const float*)d_in[18];
    const float* gnw = (const float*)d_in[19]; const float* gnb = (const float*)d_in[20];
    const float* msg_w = (const float*)d_in[21]; const float* msg_b = (const float*)d_in[22];

    int nNodes = in_sizes[0] / D;     // 50000
    int nE     = in_sizes[2];         // 800000

    float* ws = (float*)d_ws;
    float* s1 = ws + WS_BIG;
    float* s2 = s1 + nNodes;
    float* g1 = s2 + nNodes;
    float* g2 = g1 + nNodes;
    int*   segmax = (int*)(g2 + nNodes);
    float* segsum = (float*)(segmax + nNodes);
    float* VM     = segsum + nNodes;

    float* out       = (float*)d_out;
    float* out_alpha = out + (size_t)nNodes * D;   // used as e / e_exp scratch, then alpha
    float* out_gamma = out_alpha + nE;
    float* out_bij   = out_gamma + nE;

    int nTot = nNodes * D;
    init_kernel<<<(nTot + 255) / 256, 256, 0, stream>>>(out, segmax, segsum, nTot, nNodes);

    precompute_kernel<<<1, 128, 0, stream>>>(Wq, bq, Wk, bk, Wv, bv, rel_emb, attn_vec,
                                             gnw, msg_w, msg_b, ws);

    vm_gemm_kernel<<<(nNodes + 15) / 16, 256, 0, stream>>>(x, ws, VM,
                                                           s1, s2, g1, g2, nNodes);

    edge_kernel<<<(nE + 255) / 256, 256, 0, stream>>>(edge_index, edge_type, erf,
        rule_w1, rule_b1, rule_w2, rule_b2, gate_rule_w, gate_rule_b, gnb,
        ws, s1, s2, g1, g2, out_alpha, out_gamma, out_bij, segmax, nE);

    exp_sum_kernel<<<(nE + 255) / 256, 256, 0, stream>>>(edge_index, segmax,
        out_alpha, segsum, nE);

    alpha_msg_kernel<<<(nE + 7) / 8, 256, 0, stream>>>(edge_index, edge_type, VM, ws,
        segsum, out_alpha, out, nE);
}